// My_parse_cnn_layer_18992345383379
// MI455X (gfx1250) — compile-verified
//
#include <hip/hip_runtime.h>

// out[b,i,j] = 9 * sum_t g[b,j,t]*w3[i,j,t] + biasSum[i,j]
//   g[b,j,t]  = diag[b, j+t] if j+t<512 else 0,  diag[b,j] = inputs[b,j,j,0]
//   w3[i,j,t] = w[(i*512+j)*9+t] * ((i + t/3) < 512)
//   biasSum[i,j] = sum_t b[(i*512+j)*9+t]
// Bandwidth-bound (~124 MB @ 23.3 TB/s). Hot loop: v_wmma_f32_16x16x4_f32 on
// pre-masked, K-padded (9->12) operands -> all loads unconditional aligned b64;
// store guards removed by splitting full b-tiles (no guard) from the one edge
// b-tile (single kh divergence), so the steady state has no exec branches.

typedef __attribute__((ext_vector_type(2))) float v2f;
typedef __attribute__((ext_vector_type(8))) float v8f;

#define SDIM      512
#define UNITS     9
#define KPAD      12           // K padded to 3 WMMA steps of 4
#define BATCH     100
#define BTILES    7            // ceil(100/16); 6 full + 1 edge
#define ITILES    32           // 512/16
#define DIAG_W    524          // 512 + 12 zero pad
#define DIAG_ROWS 112          // 7*16, rows >= 100 zeroed
#define JCHUNK    64
#define NJCHUNKS  8            // 512/64

#define DIAG_FLOATS  (DIAG_ROWS * DIAG_W)             // 58,688
#define DIAGG_FLOATS (DIAG_ROWS * SDIM * KPAD)        // 688,128
#define WPAD_FLOATS  (SDIM * SDIM * KPAD)             // 3,145,728
#define BIAS_FLOATS  (SDIM * SDIM)                    // 262,144

// ---- prep 1: pull the diagonal out of the 105MB input (only 205KB live) ----
__global__ void prep_diag(const float* __restrict__ in, float* __restrict__ diag) {
    int idx = blockIdx.x * blockDim.x + threadIdx.x;
    if (idx >= DIAG_FLOATS) return;
    int b = idx / DIAG_W, j = idx - b * DIAG_W;
    float v = 0.0f;
    if (b < BATCH && j < SDIM)
        v = in[(size_t)b * SDIM * SDIM + (size_t)j * (SDIM + 1)];
    diag[idx] = v;
}

// ---- prep 2: diagG[b][j][12] = g[b,j,t] (zero-padded K, rows b>=100 zero) ----
__global__ void prep_diagG(const float* __restrict__ diag, float* __restrict__ diagG) {
    int idx = blockIdx.x * blockDim.x + threadIdx.x;   // b*512 + j
    if (idx >= DIAG_ROWS * SDIM) return;
    int b = idx >> 9, j = idx & (SDIM - 1);
    const float* dp = diag + (size_t)b * DIAG_W + j;   // zero-padded past 512
    float* o = diagG + (size_t)idx * KPAD;
    ((float4*)o)[0] = make_float4(dp[0], dp[1], dp[2], dp[3]);
    ((float4*)o)[1] = make_float4(dp[4], dp[5], dp[6], dp[7]);
    ((float4*)o)[2] = make_float4(dp[8], 0.0f, 0.0f, 0.0f);
}

// ---- prep 3: wPad[i][j][12] = w3 masked by (i + t/3 < 512), K zero-padded ----
__global__ void prep_wpad(const float* __restrict__ w, float* __restrict__ wPad) {
    int idx = blockIdx.x * blockDim.x + threadIdx.x;   // i*512 + j
    if (idx >= SDIM * SDIM) return;
    int i = idx >> 9;
    const float* p = w + (size_t)idx * UNITS;
    float v[KPAD];
#pragma unroll
    for (int t = 0; t < UNITS; ++t)
        v[t] = ((i + t / 3) < SDIM) ? p[t] : 0.0f;
    v[9] = v[10] = v[11] = 0.0f;
    float* o = wPad + (size_t)idx * KPAD;
    ((float4*)o)[0] = make_float4(v[0], v[1], v[2], v[3]);
    ((float4*)o)[1] = make_float4(v[4], v[5], v[6], v[7]);
    ((float4*)o)[2] = make_float4(v[8], v[9], v[10], v[11]);
}

// ---- prep 4: bias[i,j] = sum_t b3[i,j,t] (computed once, not 100x) ----
__global__ void prep_bias(const float* __restrict__ bsrc, float* __restrict__ biasPre) {
    int idx = blockIdx.x * blockDim.x + threadIdx.x;
    if (idx >= SDIM * SDIM) return;
    const float* p = bsrc + (size_t)idx * UNITS;
    float s = 0.0f;
#pragma unroll
    for (int t = 0; t < UNITS; ++t) s += p[t];
    biasPre[idx] = s;
}

// ---- main: per-j 16x16 WMMA tiles (M=b, N=i, K=t padded to 12), j grouped by
//      4 so stores are contiguous float4 on the output's fast axis.
//      FULL=true  -> b-tile fully < BATCH: unconditional stores, no branches.
//      FULL=false -> edge tile b0=96: only kh==0 lanes, rows 0..3 store. ----
template <bool FULL>
__global__ __launch_bounds__(256)
void cnn_diag_wmma(const float* __restrict__ wPad, const float* __restrict__ diagG,
                   const float* __restrict__ biasPre, float* __restrict__ out) {
    const int lane  = threadIdx.x & 31;
    const int wid   = blockIdx.x * 8 + (threadIdx.x >> 5);
    const int b_tile = FULL ? (wid >> 8) : 6;     // FULL: 0..5
    const int rem    = wid & 255;
    const int i_tile = rem >> 3;                  // 0..31
    const int chunk  = rem & 7;                   // 0..7
    const int b0 = b_tile * 16, i0 = i_tile * 16, j0 = chunk * JCHUNK;
    const int n  = lane & 15;                     // tile row (A: M=b) / col (B: N=i)
    const int kh = lane >> 4;                     // K-half this lane feeds
    const int iB = i0 + n;

    // lane's K offsets per WMMA step s (K base 4s): supplies t0, t0+1
    const int t0s[3] = {2 * kh, 4 + 2 * kh, 8 + 2 * kh};

    const float* dBase = diagG + (size_t)(b0 + n) * (SDIM * KPAD);
    const float* wBase = wPad  + (size_t)iB * (SDIM * KPAD);
    float* outRow = out + (size_t)iB * SDIM;      // + b*SDIM*SDIM + j later

    for (int jg = 0; jg < JCHUNK; jg += 4) {
        const int j = j0 + jg;
        const float4 bias4 = *(const float4*)(biasPre + (size_t)iB * SDIM + j);
        v8f c[4];
#pragma unroll
        for (int q = 0; q < 4; ++q) {
            const float* dp = dBase + (size_t)(j + q) * KPAD;
            const float* wp = wBase + (size_t)(j + q) * KPAD;
            v8f acc = {};
#pragma unroll
            for (int s = 0; s < 3; ++s) {
                v2f a  = *(const v2f*)(dp + t0s[s]);   // aligned, pre-masked
                v2f bb = *(const v2f*)(wp + t0s[s]);   // aligned, pre-masked
                acc = __builtin_amdgcn_wmma_f32_16x16x4_f32(
                    false, a, false, bb, (short)0, acc, false, false);
            }
            c[q] = acc;
        }
        // D layout: lane L, VGPR r -> M = r + 8*(L>>4), N = L&15
        if (FULL) {
#pragma unroll
            for (int r = 0; r < 8; ++r) {
                const int bb_ = b0 + r + 8 * kh;       // <= 95 < 100 always
                float4 o;
                o.x = 9.0f * c[0][r] + bias4.x;
                o.y = 9.0f * c[1][r] + bias4.y;
                o.z = 9.0f * c[2][r] + bias4.z;
                o.w = 9.0f * c[3][r] + bias4.w;
                *(float4*)(outRow + (size_t)bb_ * (SDIM * SDIM) + j) = o;
            }
        } else {
            if (kh == 0) {                             // rows 96..99 only
#pragma unroll
                for (int r = 0; r < 4; ++r) {
                    const int bb_ = b0 + r;
                    float4 o;
                    o.x = 9.0f * c[0][r] + bias4.x;
                    o.y = 9.0f * c[1][r] + bias4.y;
                    o.z = 9.0f * c[2][r] + bias4.z;
                    o.w = 9.0f * c[3][r] + bias4.w;
                    *(float4*)(outRow + (size_t)bb_ * (SDIM * SDIM) + j) = o;
                }
            }
        }
    }
}

// ---- fallback (only if ws too small): plain coalesced VALU kernel ----
__global__ __launch_bounds__(256)
void cnn_diag_fallback(const float* __restrict__ in, const float* __restrict__ w,
                       const float* __restrict__ bsrc, float* __restrict__ out) {
    size_t o = (size_t)blockIdx.x * blockDim.x + threadIdx.x;
    if (o >= (size_t)BATCH * SDIM * SDIM) return;
    int b = (int)(o >> 18);
    int r = (int)(o & (SDIM * SDIM - 1));
    int i = r >> 9, j = r & (SDIM - 1);
    const float* wp = w + (size_t)r * UNITS;
    const float* bp = bsrc + (size_t)r * UNITS;
    float dot = 0.0f, bias = 0.0f;
#pragma unroll
    for (int t = 0; t < UNITS; ++t) {
        float g = ((j + t) < SDIM)
                    ? in[(size_t)b * SDIM * SDIM + (size_t)(j + t) * (SDIM + 1)] : 0.0f;
        float wv = ((i + t / 3) < SDIM) ? wp[t] : 0.0f;
        dot += g * wv;
        bias += bp[t];
    }
    out[o] = 9.0f * dot + bias;
}

extern "C" void kernel_launch(void* const* d_in, const int* in_sizes, int n_in,
                              void* d_out, int out_size, void* d_ws, size_t ws_size,
                              hipStream_t stream) {
    const float* inputs = (const float*)d_in[0];
    const float* w      = (const float*)d_in[1];
    const float* b      = (const float*)d_in[2];
    float* out = (float*)d_out;

    const size_t needF = (size_t)DIAG_FLOATS + DIAGG_FLOATS + WPAD_FLOATS + BIAS_FLOATS;
    if (ws_size >= needF * sizeof(float)) {
        float* diag    = (float*)d_ws;                  // all sections 16B-aligned
        float* diagG   = diag  + DIAG_FLOATS;
        float* wPad    = diagG + DIAGG_FLOATS;
        float* biasPre = wPad  + WPAD_FLOATS;

        prep_diag <<<(DIAG_FLOATS + 255) / 256, 256, 0, stream>>>(inputs, diag);
        prep_diagG<<<(DIAG_ROWS * SDIM + 255) / 256, 256, 0, stream>>>(diag, diagG);
        prep_wpad <<<(SDIM * SDIM + 255) / 256, 256, 0, stream>>>(w, wPad);
        prep_bias <<<(SDIM * SDIM + 255) / 256, 256, 0, stream>>>(b, biasPre);

        // 6 full b-tiles: 6*32*8 = 1536 waves = 192 blocks; edge tile: 32 blocks
        cnn_diag_wmma<true ><<<192, 256, 0, stream>>>(wPad, diagG, biasPre, out);
        cnn_diag_wmma<false><<< 32, 256, 0, stream>>>(wPad, diagG, biasPre, out);
    } else {
        size_t nTot = (size_t)BATCH * SDIM * SDIM;
        cnn_diag_fallback<<<(unsigned)((nTot + 255) / 256), 256, 0, stream>>>(inputs, w, b, out);
    }
}